// HarmSynth_15710990369693
// MI455X (gfx1250) — compile-verified
//
#include <hip/hip_runtime.h>
#include <math.h>

#define NB_HARMS   64
#define B_SZ       16
#define L_SZ       512
#define T_SZ       65536
#define CHUNKSZ    512
#define NCHUNK     128          // T / 512
#define NTILE      4096         // T / 16
#define NBLK       40           // ceil(511 / BLK_STEP)
#define BLK_STEP   13           // frames of i0 per block (16-frame window -> 2 slack)

#define TWO_PI_F   6.28318530717958647692f
#define SR_F       16000.0f
#define NYQ_F      8000.0f

typedef float v2f __attribute__((ext_vector_type(2)));
typedef float v8f __attribute__((ext_vector_type(8)));

// pos = t * (L-1)/(T-1) in f32, i0 = clip(floor(pos),0,510), w = pos - i0  (matches reference)
__device__ __forceinline__ void pos_of(int t, int* i0o, float* wo) {
    const float step = (float)(511.0 / 65535.0);
    float pos = (float)t * step;
    int i0 = (int)floorf(pos);
    i0 = i0 < 0 ? 0 : (i0 > 510 ? 510 : i0);
    *i0o = i0;
    *wo  = pos - (float)i0;
}

__device__ __forceinline__ float phi_at(const float* f0b, int t) {
    int i0; float w;
    pos_of(t, &i0, &w);
    float f0i = f0b[i0] * (1.0f - w) + f0b[i0 + 1] * w;
    return fmodf(TWO_PI_F * f0i / SR_F, TWO_PI_F);
}

// ---------- pass 0: mask above Nyquist, normalize over H, * a0 -> aaN[b][h][l] ----------
__global__ void k_prep_amp(const float* __restrict__ f0, const float* __restrict__ a0,
                           const float* __restrict__ aa, float* __restrict__ aaN) {
    int idx = blockIdx.x * blockDim.x + threadIdx.x;   // b*L + l
    if (idx >= B_SZ * L_SZ) return;
    int b = idx / L_SZ, l = idx % L_SZ;
    float f = f0[idx];
    float sum = 0.f;
    for (int h = 0; h < NB_HARMS; ++h) {
        float ff = f * (float)(h + 1);
        float v  = aa[(b * NB_HARMS + h) * L_SZ + l];
        sum += (ff >= NYQ_F) ? 0.f : v;
    }
    float scale = a0[idx] / sum;                       // sum==0 -> inf, same as reference
    for (int h = 0; h < NB_HARMS; ++h) {
        float ff = f * (float)(h + 1);
        float v  = aa[(b * NB_HARMS + h) * L_SZ + l];
        v = (ff >= NYQ_F) ? 0.f : v;
        aaN[(b * NB_HARMS + h) * L_SZ + l] = v * scale;
    }
}

// ---------- pass 1: per-16-sample group sums of phase ----------
__global__ void k_group_sums(const float* __restrict__ f0, float* __restrict__ G) {
    int idx = blockIdx.x * blockDim.x + threadIdx.x;   // b*NTILE + j
    if (idx >= B_SZ * NTILE) return;
    int b = idx / NTILE, j = idx % NTILE;
    const float* f0b = f0 + b * L_SZ;
    float s = 0.f;
    for (int d = 0; d < 16; ++d) s += phi_at(f0b, 16 * j + d);
    G[idx] = s;
}

// ---------- pass 2: chunk-carry scan + per-tile exclusive prefixes ----------
// theta(t) = (c_k - u_k) + Q[j] + inclusive_scan_in_tile(phi)   with u_k = phi(512k)
__global__ void k_scan(const float* __restrict__ f0, const float* __restrict__ G,
                       float* __restrict__ Q, float* __restrict__ cmu) {
    int b = blockIdx.x;
    int lane = threadIdx.x;                            // 32 lanes, one wave
    const float* f0b = f0 + b * L_SZ;
    float s = 0.f;
    for (int k = 0; k < NCHUNK; ++k) {
        float g = G[b * NTILE + k * 32 + lane];
        float x = g;
        for (int off = 1; off < 32; off <<= 1) {
            float y = __shfl_up(x, off, 32);
            if (lane >= off) x += y;
        }
        Q[b * NTILE + k * 32 + lane] = x - g;          // exclusive prefix within chunk
        float total = __shfl(x, 31, 32);
        float u = phi_at(f0b, k * CHUNKSZ);
        float c = fmodf(s + u, TWO_PI_F);              // carry with mod at chunk start
        if (lane == 0) cmu[b * NCHUNK + k] = c - u;
        s = c + (total - u);
    }
}

// ---------- pass 3: main synth; WMMA harmonic reduction ----------
__global__ __launch_bounds__(256) void k_synth(const float* __restrict__ f0,
                                               const float* __restrict__ aaN,
                                               const float* __restrict__ Q,
                                               const float* __restrict__ cmu,
                                               float* __restrict__ out) {
    __shared__ float sh[8][256];                       // per-wave 16x16 D spill
    const int p    = blockIdx.x;                       // frame block
    const int b    = blockIdx.y;                       // batch
    const int lane = threadIdx.x & 31;
    const int wid  = threadIdx.x >> 5;
    const int nw   = blockDim.x >> 5;
    const int m    = lane & 15;
    const bool hi  = lane >= 16;

    // tiles j with floor(i0(16j)/BLK_STEP) == p : i0(16j) = floor(8176j/65535)
    long long jlo = ((long long)65535 * BLK_STEP * p + 8175) / 8176;
    long long jhi = ((long long)65535 * BLK_STEP * (p + 1) + 8175) / 8176 - 1;
    if (jhi > NTILE - 1) jhi = NTILE - 1;
    if (jlo > jhi) return;                             // uniform over block
    const int l0 = BLK_STEP * p;

    // A operand: aaN^T frame window, M=l_local(16) x K=h(64), static per block.
    // Layout per 4-K group jg: lane<16 holds K=4jg,4jg+1 ; lane>=16 holds K=4jg+2,4jg+3 ; M=lane%16.
    float a0v[16], a1v[16];
    {
        const float* ab = aaN + b * NB_HARMS * L_SZ;
        int col = l0 + m; if (col > L_SZ - 1) col = L_SZ - 1;   // clamp (tail rows unused)
#pragma unroll
        for (int jg = 0; jg < 16; ++jg) {
            int k0 = 4 * jg + (hi ? 2 : 0);
            a0v[jg] = ab[ k0      * L_SZ + col];
            a1v[jg] = ab[(k0 + 1) * L_SZ + col];
        }
    }

    const float* f0b = f0 + b * L_SZ;
    for (int j = (int)jlo + wid; j <= (int)jhi; j += nw) {
        int t = 16 * j + m;                            // lanes 16-31 duplicate t of lanes 0-15
        int k = t >> 9;                                // chunk id (tile never crosses a chunk)
        int i0; float w;
        pos_of(t, &i0, &w);
        float f0i = f0b[i0] * (1.0f - w) + f0b[i0 + 1] * w;
        float phi = fmodf(TWO_PI_F * f0i / SR_F, TWO_PI_F);

        // inclusive scan of phi over 16-lane segments (both halves identical)
        float x = phi;
#pragma unroll
        for (int off = 1; off < 16; off <<= 1) {
            float y = __shfl_up(x, off, 16);
            if ((lane & 15) >= off) x += y;
        }
        float theta = cmu[b * NCHUNK + k] + Q[b * NTILE + j] + x;
        theta = fmodf(theta, TWO_PI_F);

        // sin(1..4)theta and 2*cos(4theta) for stride-4 Chebyshev chains
        float s1, c1;
        __sincosf(theta, &s1, &c1);
        float twoC1 = 2.0f * c1;
        float s2 = s1 * twoC1;                               // sin 2t
        float s3 = __builtin_fmaf(twoC1, s2, -s1);           // sin 3t
        float s4 = __builtin_fmaf(twoC1, s3, -s2);           // sin 4t
        float c4 = __builtin_fmaf(-2.0f * s2, s2, 1.0f);     // cos 4t = 1 - 2 sin^2 2t
        float twoC4 = 2.0f * c4;

        // Two chains per lane, stepping ranks by 4:
        //  lane<16 : b0 ranks 1,5,9,..  (prev -s3, cur s1) ; b1 ranks 2,6,10,.. (prev -s2, cur s2)
        //  lane>=16: b0 ranks 3,7,11,.. (prev -s1, cur s3) ; b1 ranks 4,8,12,.. (prev 0,   cur s4)
        float p0 = hi ? -s1 : -s3;
        float q0 = hi ?  s3 :  s1;
        float p1 = hi ?  0.f : -s2;
        float q1 = hi ?  s4 :  s2;

        v8f d = {0.f, 0.f, 0.f, 0.f, 0.f, 0.f, 0.f, 0.f};
#pragma unroll
        for (int jg = 0; jg < 16; ++jg) {
            v2f av = { a0v[jg], a1v[jg] };
            v2f bv = { q0, q1 };
            // D[l_local, t] += A(16x4) x B(4x16)
            d = __builtin_amdgcn_wmma_f32_16x16x4_f32(false, av, false, bv,
                                                      (short)0, d, false, false);
            float n0 = __builtin_fmaf(twoC4, q0, -p0); p0 = q0; q0 = n0;
            float n1 = __builtin_fmaf(twoC4, q1, -p1); p1 = q1; q1 = n1;
        }

        // spill D (M=frame row, N=t col) through LDS; gather 2 frame rows per sample
        float* s = sh[wid];
#pragma unroll
        for (int r = 0; r < 8; ++r)
            s[(r + (hi ? 8 : 0)) * 16 + m] = d[r];
        int fi = i0 - l0; fi = fi < 0 ? 0 : (fi > 14 ? 14 : fi);
        float d0 = s[ fi      * 16 + m];
        float d1 = s[(fi + 1) * 16 + m];
        float o  = d0 + w * (d1 - d0);
        if (!hi) out[b * T_SZ + t] = o;
    }
}

extern "C" void kernel_launch(void* const* d_in, const int* in_sizes, int n_in,
                              void* d_out, int out_size, void* d_ws, size_t ws_size,
                              hipStream_t stream) {
    const float* f0 = (const float*)d_in[0];   // [16,512]
    const float* a0 = (const float*)d_in[1];   // [16,512]
    const float* aa = (const float*)d_in[2];   // [16,64,512]
    float* out = (float*)d_out;                // [16,65536]

    float* ws  = (float*)d_ws;
    float* aaN = ws;                                   // 16*64*512 = 524288
    float* G   = aaN + B_SZ * NB_HARMS * L_SZ;         // 16*4096   =  65536
    float* Q   = G   + B_SZ * NTILE;                   // 16*4096   =  65536
    float* cmu = Q   + B_SZ * NTILE;                   // 16*128    =   2048
    (void)in_sizes; (void)n_in; (void)out_size; (void)ws_size;

    k_prep_amp  <<<(B_SZ * L_SZ + 255) / 256, 256, 0, stream>>>(f0, a0, aa, aaN);
    k_group_sums<<<(B_SZ * NTILE + 255) / 256, 256, 0, stream>>>(f0, G);
    k_scan      <<<B_SZ, 32, 0, stream>>>(f0, G, Q, cmu);
    k_synth     <<<dim3(NBLK, B_SZ), 256, 0, stream>>>(f0, aaN, Q, cmu, out);
}